// RNN_52913997087122
// MI455X (gfx1250) — compile-verified
//
#include <hip/hip_runtime.h>

// ---------------------------------------------------------------------------
// Fused tanh-RNN scan for MI455X (gfx1250, wave32, WMMA).
//
// Shapes: B=4096, T=2048, I=16, H=20 (padded to 32 = two WMMA N-tiles).
// One wave (32 threads) owns a 16-batch tile and runs the whole T-scan.
// Software pipeline: xp_{t+1} = bias + x_{t+1} W_ih^T is accumulated in an
// independent WMMA chain while the serial part of step t runs:
//   h_{t+1} = tanh(xp_t + h_t W_hh^T)   (5 chained v_wmma_f32_16x16x4_f32)
// D (N-striped-across-lanes) -> LDS -> reload as A (K-striped-across-VGPRs).
// ---------------------------------------------------------------------------

typedef float v2f __attribute__((ext_vector_type(2)));
typedef float v8f __attribute__((ext_vector_type(8)));

#define TT 2048
#define BV 4096
#define II 16
#define HH 20
#define LSTRIDE 34   // LDS row stride in floats: 8B-aligned, bank-spread

__device__ __forceinline__ float fast_tanh(float v) {
#if __has_builtin(__builtin_amdgcn_tanhf)
    return __builtin_amdgcn_tanhf(v);          // v_tanh_f32 (gfx1250 trans op)
#else
    v = fminf(fmaxf(v, -9.0f), 9.0f);
    float e = __builtin_amdgcn_exp2f(v * -2.8853900817779268f); // 2*log2(e)
    return (1.0f - e) * __builtin_amdgcn_rcpf(1.0f + e);
#endif
}

__global__ __launch_bounds__(32) void rnn_wmma_kernel(
    const float* __restrict__ x,    // [B, T, I]
    const float* __restrict__ Wih,  // [H, I]
    const float* __restrict__ Whh,  // [H, H]
    const float* __restrict__ bih,  // [H]
    const float* __restrict__ bhh,  // [H]
    const float* __restrict__ Wfc,  // [O=1, H]
    const float* __restrict__ bfc,  // [O=1]
    float* __restrict__ out)        // [B, 1]
{
    __shared__ __align__(16) float hl[16 * LSTRIDE];

    const int lane = threadIdx.x;   // 0..31
    const int half = lane >> 4;     // 0/1: selects K-sub-pair in A/B layout
    const int lp   = lane & 15;     // M (batch row) for A, N for B/C/D
    const int bb   = blockIdx.x * 16;

    // ---- B-matrix layout weight staging (K=4 chunks, N=16 tiles) ----------
    // B[k][n]: VGPR v holds K = 4c + 2*half + v, N = lp.  Rows j>=H are 0,
    // so padded h stays exactly tanh(0)=0 and never perturbs the recurrence.
    v2f wih[2][4];
#pragma unroll
    for (int m = 0; m < 2; ++m) {
        const int j = 16 * m + lp;
#pragma unroll
        for (int c = 0; c < 4; ++c) {
            v2f w = {0.f, 0.f};
            if (j < HH) w = *(const v2f*)(Wih + j * II + 4 * c + 2 * half);
            wih[m][c] = w;
        }
    }
    v2f whh[2][5];
#pragma unroll
    for (int m = 0; m < 2; ++m) {
        const int j = 16 * m + lp;
#pragma unroll
        for (int c = 0; c < 5; ++c) {
            v2f w = {0.f, 0.f};
            if (j < HH) w = *(const v2f*)(Whh + j * HH + 4 * c + 2 * half);
            whh[m][c] = w;
        }
    }
    float bias[2];
#pragma unroll
    for (int m = 0; m < 2; ++m) {
        const int j = 16 * m + lp;
        bias[m] = (j < HH) ? (bih[j] + bhh[j]) : 0.f;
    }

    // Per-lane x stream base: A layout wants (b=lp, k=4c+2*half+{0,1}).
    const float* px = x + (size_t)(bb + lp) * TT * II + 2 * half;

    // ---- prologue: xp_0 = bias + x_0 W_ih^T  (independent WMMA chain) -----
    v8f xp0, xp1;
#pragma unroll
    for (int i = 0; i < 8; ++i) { xp0[i] = bias[0]; xp1[i] = bias[1]; }
    {
        v2f xv[4];
#pragma unroll
        for (int c = 0; c < 4; ++c) xv[c] = *(const v2f*)(px + 4 * c);
#pragma unroll
        for (int c = 0; c < 4; ++c) {
            xp0 = __builtin_amdgcn_wmma_f32_16x16x4_f32(false, xv[c], false, wih[0][c],
                                                        (short)0, xp0, false, false);
            xp1 = __builtin_amdgcn_wmma_f32_16x16x4_f32(false, xv[c], false, wih[1][c],
                                                        (short)0, xp1, false, false);
        }
    }

    // h_0 = 0 in A-chunk layout.
    v2f ha[5];
#pragma unroll
    for (int c = 0; c < 5; ++c) ha[c] = (v2f){0.f, 0.f};

    // ---- main scan: iterations t = 0..TT-2 produce h_{t+1} and xp_{t+1} ---
#pragma unroll 2
    for (int t = 0; t < TT - 1; ++t) {
        // Next timestep's x tile (independent of the recurrence chain).
        v2f xn[4];
#pragma unroll
        for (int c = 0; c < 4; ++c)
            xn[c] = *(const v2f*)(px + (size_t)(t + 1) * II + 4 * c);
        if (t + 8 < TT)
            __builtin_prefetch(px + (size_t)(t + 8) * II, 0, 1); // global_prefetch_b8

        // Serial part: D = xp_t + h_t W_hh^T, 5 chained K=4 WMMAs per N-tile.
        v8f d0 = xp0, d1 = xp1;
#pragma unroll
        for (int c = 0; c < 5; ++c) {
            d0 = __builtin_amdgcn_wmma_f32_16x16x4_f32(false, ha[c], false, whh[0][c],
                                                       (short)0, d0, false, false);
            d1 = __builtin_amdgcn_wmma_f32_16x16x4_f32(false, ha[c], false, whh[1][c],
                                                       (short)0, d1, false, false);
        }

        // Overlapped part: xp_{t+1} = bias + x_{t+1} W_ih^T (own accumulator).
#pragma unroll
        for (int i = 0; i < 8; ++i) { xp0[i] = bias[0]; xp1[i] = bias[1]; }
#pragma unroll
        for (int c = 0; c < 4; ++c) {
            xp0 = __builtin_amdgcn_wmma_f32_16x16x4_f32(false, xn[c], false, wih[0][c],
                                                        (short)0, xp0, false, false);
            xp1 = __builtin_amdgcn_wmma_f32_16x16x4_f32(false, xn[c], false, wih[1][c],
                                                        (short)0, xp1, false, false);
        }

        // Nonlinearity (rows j>=H stay tanh(0)=0 by construction).
#pragma unroll
        for (int i = 0; i < 8; ++i) { d0[i] = fast_tanh(d0[i]); d1[i] = fast_tanh(d1[i]); }

        // D layout -> LDS [b][j]: VGPR r holds (b = r + 8*half, j = lp | 16+lp).
        __syncthreads();   // single-wave WG: near-NOP barrier, compiler ordering
#pragma unroll
        for (int r = 0; r < 8; ++r) {
            hl[(r + 8 * half) * LSTRIDE + lp]      = d0[r];
            hl[(r + 8 * half) * LSTRIDE + 16 + lp] = d1[r];
        }
        __syncthreads();
        // Reload as A-chunks: lane lp = batch row, K pair = 4c + 2*half.
#pragma unroll
        for (int c = 0; c < 5; ++c)
            ha[c] = *(const v2f*)(&hl[lp * LSTRIDE + 4 * c + 2 * half]);
    }

    // ---- peeled final step (t = TT-1): h_T = tanh(xp_{T-1} + h Whh^T) -----
    {
        v8f d0 = xp0, d1 = xp1;
#pragma unroll
        for (int c = 0; c < 5; ++c) {
            d0 = __builtin_amdgcn_wmma_f32_16x16x4_f32(false, ha[c], false, whh[0][c],
                                                       (short)0, d0, false, false);
            d1 = __builtin_amdgcn_wmma_f32_16x16x4_f32(false, ha[c], false, whh[1][c],
                                                       (short)0, d1, false, false);
        }
#pragma unroll
        for (int i = 0; i < 8; ++i) { d0[i] = fast_tanh(d0[i]); d1[i] = fast_tanh(d1[i]); }
        __syncthreads();
#pragma unroll
        for (int r = 0; r < 8; ++r) {
            hl[(r + 8 * half) * LSTRIDE + lp]      = d0[r];
            hl[(r + 8 * half) * LSTRIDE + 16 + lp] = d1[r];
        }
        __syncthreads();
    }

    // ---- final fc: out[b] = sum_j h_T[b,j] * Wfc[j] + bfc ------------------
    if (lane < 16) {
        float acc = bfc[0];
#pragma unroll
        for (int j = 0; j < HH; ++j)
            acc += hl[lane * LSTRIDE + j] * Wfc[j];
        out[bb + lane] = acc;
    }
}

extern "C" void kernel_launch(void* const* d_in, const int* in_sizes, int n_in,
                              void* d_out, int out_size, void* d_ws, size_t ws_size,
                              hipStream_t stream) {
    (void)in_sizes; (void)n_in; (void)out_size; (void)d_ws; (void)ws_size;
    const float* x   = (const float*)d_in[0];
    const float* Wih = (const float*)d_in[1];
    const float* Whh = (const float*)d_in[2];
    const float* bih = (const float*)d_in[3];
    const float* bhh = (const float*)d_in[4];
    const float* Wfc = (const float*)d_in[5];
    const float* bfc = (const float*)d_in[6];

    rnn_wmma_kernel<<<BV / 16, 32, 0, stream>>>(x, Wih, Whh, bih, bhh, Wfc, bfc,
                                                (float*)d_out);
}